// ROIAlignRotated3D_57045755625760
// MI455X (gfx1250) — compile-verified
//
#include <hip/hip_runtime.h>
#include <hip/hip_bf16.h>
#include <math.h>

typedef __attribute__((ext_vector_type(2))) float v2f;
typedef __attribute__((ext_vector_type(8))) float v8f;

#define POOLED 7
#define NBINS  49          // 7*7
#define NSAMP  4           // 2x2 sampling grid
#define C_     256
#define H_     200
#define W_     176
#define HW_    (H_ * W_)
#define CHW_   (C_ * HW_)
#define SCALE  0.25f

// One block per ROI. 256 threads = 8 waves (wave32).
// Stage 1: threads 0..195 compute the 49*4 sample records (4 weights, 4 BYTE offsets) into LDS.
// Stage 2: each wave owns 2 channel-groups of 16 channels; per (group, bin) it gathers
//          16 taps x 16 channels and reduces with 4x V_WMMA_F32_16X16X4_F32.
//          Feature base is scalar-uniform (readfirstlane'd batch) so gathers use
//          SADDR + 32-bit VGPR byte-offset addressing.
__global__ __launch_bounds__(256) void roi_align_rotated_wmma(
    const float* __restrict__ feat,
    const float* __restrict__ rois0,
    float* __restrict__ out)
{
    __shared__ float sw[NBINS * NSAMP * 4];   // weights (valid mask & 1/4 mean folded in)
    __shared__ int   soff[NBINS * NSAMP * 4]; // byte offsets within a channel plane

    const int n = blockIdx.x;
    const int t = threadIdx.x;

    const float* roi = rois0 + n * 8;
    const int   b     = (int)roi[0];
    const float chc   = roi[1] * SCALE - 0.5f;   // center h
    const float cwc   = roi[2] * SCALE - 0.5f;   // center w
    const float rh    = roi[4] * SCALE;
    const float rw    = roi[5] * SCALE;
    const float theta = roi[7];

    if (t < NBINS * NSAMP) {
        const int bin = t >> 2;
        const int s   = t & 3;
        const int ph  = bin / POOLED;
        const int pw  = bin % POOLED;
        const int iy  = s >> 1;
        const int ix  = s & 1;

        const float bin_h = rh * (1.0f / POOLED);
        const float bin_w = rw * (1.0f / POOLED);
        const float yy = -0.5f * rh + (float)ph * bin_h + ((float)iy + 0.5f) * bin_h * 0.5f;
        const float xx = -0.5f * rw + (float)pw * bin_w + ((float)ix + 0.5f) * bin_w * 0.5f;

        float sinT, cosT;
        __sincosf(theta, &sinT, &cosT);
        const float y = yy * cosT - xx * sinT + chc;
        const float x = yy * sinT + xx * cosT + cwc;

        const bool valid = (y >= -1.0f) && (y <= (float)H_) &&
                           (x >= -1.0f) && (x <= (float)W_);

        float yc = fmaxf(y, 0.0f);
        float xc = fmaxf(x, 0.0f);
        int y0 = (int)floorf(yc);
        int x0 = (int)floorf(xc);
        if (y0 >= H_ - 1) { yc = (float)(H_ - 1); y0 = H_ - 1; }
        if (x0 >= W_ - 1) { xc = (float)(W_ - 1); x0 = W_ - 1; }
        const int y1 = min(y0 + 1, H_ - 1);
        const int x1 = min(x0 + 1, W_ - 1);

        const float ly = yc - (float)y0;
        const float lx = xc - (float)x0;
        const float hy = 1.0f - ly;
        const float hx = 1.0f - lx;
        const float vm = valid ? 0.25f : 0.0f;   // fold 2x2-mean + validity into weights

        const int base = t * 4;
        sw[base + 0] = hy * hx * vm;  soff[base + 0] = (y0 * W_ + x0) * 4;
        sw[base + 1] = hy * lx * vm;  soff[base + 1] = (y0 * W_ + x1) * 4;
        sw[base + 2] = ly * hx * vm;  soff[base + 2] = (y1 * W_ + x0) * 4;
        sw[base + 3] = ly * lx * vm;  soff[base + 3] = (y1 * W_ + x1) * 4;
    }
    __syncthreads();

    const int lane = t & 31;
    const int wave = t >> 5;
    const int m    = lane & 15;   // channel within 16-channel group (A-matrix M index)
    const int h    = lane >> 4;   // K-half: this lane supplies taps {2h, 2h+1} of each chunk

    // Batch index is block-uniform; force it scalar so the 64-bit base stays in SGPRs
    // and per-gather addressing is SADDR + zext(32-bit VGPR byte offset).
    const int bu = __builtin_amdgcn_readfirstlane(b);
    const char* __restrict__ fb0 = (const char*)(feat + (size_t)bu * CHW_);
    float* outn = out + (size_t)n * (C_ * NBINS);

    #pragma unroll
    for (int cc = 0; cc < 2; ++cc) {
        const int cg = wave * 2 + cc;                               // channel group 0..15
        const unsigned chOff = (unsigned)((cg * 16 + m) * HW_) * 4u; // per-lane plane byte offset

        for (int bin = 0; bin < NBINS; ++bin) {
            v8f acc = {0.f, 0.f, 0.f, 0.f, 0.f, 0.f, 0.f, 0.f};

            #pragma unroll
            for (int s = 0; s < NSAMP; ++s) {
                // taps j = 4*s + {2h, 2h+1}: this lane's A slice and B slice
                const int rec = (bin * NSAMP + s) * 4 + 2 * h;
                const float w0 = sw[rec];
                const float w1 = sw[rec + 1];
                const unsigned o0 = (unsigned)soff[rec];
                const unsigned o1 = (unsigned)soff[rec + 1];

                const float g0 = *(const float*)(fb0 + (size_t)(chOff + o0));
                const float g1 = *(const float*)(fb0 + (size_t)(chOff + o1));

                v2f a; a.x = g0; a.y = g1;          // A(16x4) chunk
                v2f w; w.x = w0; w.y = w1;          // B(4x16) chunk (N-replicated)

                acc = __builtin_amdgcn_wmma_f32_16x16x4_f32(
                    /*neg_a=*/false, a, /*neg_b=*/false, w,
                    /*c_mod=*/(short)0, acc,
                    /*reuse_a=*/false, /*reuse_b=*/false);
            }

            // D columns are identical; lane 0 holds channels 0..7 of the group in
            // acc[0..7], lane 16 holds channels 8..15. Each writes 8 strided floats.
            if (m == 0) {
                float* o = outn + (size_t)(cg * 16 + h * 8) * NBINS + bin;
                o[0 * NBINS] = acc[0];
                o[1 * NBINS] = acc[1];
                o[2 * NBINS] = acc[2];
                o[3 * NBINS] = acc[3];
                o[4 * NBINS] = acc[4];
                o[5 * NBINS] = acc[5];
                o[6 * NBINS] = acc[6];
                o[7 * NBINS] = acc[7];
            }
        }
    }
}

extern "C" void kernel_launch(void* const* d_in, const int* in_sizes, int n_in,
                              void* d_out, int out_size, void* d_ws, size_t ws_size,
                              hipStream_t stream) {
    const float* feat  = (const float*)d_in[0];
    const float* rois0 = (const float*)d_in[1];
    float*       out   = (float*)d_out;
    const int n_rois = in_sizes[1] / 8;   // rois0 is (N, 8)

    roi_align_rotated_wmma<<<dim3(n_rois), dim3(256), 0, stream>>>(feat, rois0, out);
}